// AttnModel_44994077393331
// MI455X (gfx1250) — compile-verified
//
#include <hip/hip_runtime.h>
#include <math.h>

typedef __attribute__((ext_vector_type(2))) float v2f;
typedef __attribute__((ext_vector_type(4))) float v4f;
typedef __attribute__((ext_vector_type(8))) float v8f;

#define DIMD      256
#define NB        16
#define NO        64
#define NI        512
#define NEG_SLOPE 0.01f

// CDNA5 has V_TANH_F32 on the TRANS path; use it if clang exposes the builtin.
__device__ __forceinline__ float fast_tanh(float x) {
#if __has_builtin(__builtin_amdgcn_tanhf)
  return __builtin_amdgcn_tanhf(x);
#elif __has_builtin(__builtin_amdgcn_tanh_f32)
  return __builtin_amdgcn_tanh_f32(x);
#else
  // tanh(x) = 1 - 2/(exp(2x)+1): one v_exp_f32 + cheap VALU
  return 1.0f - 2.0f / (__expf(2.0f * x) + 1.0f);
#endif
}

// C[r][c] = act( sum_d A[r*256+d] * W[c*ldW + off + d] + bias[c] )
// One wave per 16x16 C tile, K-loop of V_WMMA_F32_16X16X4_F32.
// A-frag (16x4 f32): lane<16 holds A[M=lane][K=0,1], lane>=16 holds A[M=lane-16][K=2,3].
// B-frag (4x16): B[d][c] = W[c*ldW+off+d] -> same contiguous-v2f addressing.
// C/D (16x16 f32): VGPR r, lanes 0-15 -> M=r, lanes 16-31 -> M=r+8, N=lane&15.
__global__ void wmma_gemm_kernel(const float* __restrict__ A,
                                 const float* __restrict__ W,
                                 const float* __restrict__ bias,
                                 float* __restrict__ C,
                                 int ldW, int off, int nRowTiles, int leaky) {
  const int wave = threadIdx.x >> 5;
  const int lane = threadIdx.x & 31;
  const int tile = blockIdx.x * (blockDim.x >> 5) + wave;
  const int nTiles = nRowTiles * (DIMD / 16);
  if (tile >= nTiles) return;  // wave-uniform: EXEC stays all-ones for WMMA

  const int rt = tile >> 4;          // 16 column tiles (256/16)
  const int ct = tile & 15;
  const int r0 = rt << 4;
  const int c0 = ct << 4;
  const int m    = lane & 15;        // row of A-frag / col of B-frag
  const int kk   = (lane >> 4) << 1; // K sub-offset: 0 or 2
  const int half = lane >> 4;

  const float* Ap = A + (size_t)(r0 + m) * DIMD + kk;
  const float* Wp = W + (size_t)(c0 + m) * ldW + off + kk;

  v8f acc = {0.f, 0.f, 0.f, 0.f, 0.f, 0.f, 0.f, 0.f};
#pragma unroll 8
  for (int d0 = 0; d0 < DIMD; d0 += 4) {
    v2f a = *(const v2f*)(Ap + d0);
    v2f b = *(const v2f*)(Wp + d0);
    acc = __builtin_amdgcn_wmma_f32_16x16x4_f32(
        /*neg_a=*/false, a, /*neg_b=*/false, b,
        /*c_mod=*/(short)0, acc, /*reuse_a=*/false, /*reuse_b=*/false);
  }

  const int col = c0 + m;
  const float bv = bias ? bias[col] : 0.0f;
#pragma unroll
  for (int r = 0; r < 8; ++r) {
    const int row = r0 + r + (half << 3);
    float val = acc[r] + bv;
    if (leaky) val = (val >= 0.0f) ? val : NEG_SLOPE * val;
    C[(size_t)row * DIMD + col] = val;
  }
}

// One 256-thread block (8 waves) per (b,o):
//   phase 1: score[i] = sum_h W2[h]*tanh(qp[h]+kp[b,i,h]+b1[h]) + b2, mask -> -inf
//   phase 2: softmax over i=512 (block reduction), write attn to global
//   phase 3: outA[b,o,d] = sum_i attn[i]*v[b,i,d]  (coalesced over d=tid)
__global__ void score_softmax_av_kernel(const float* __restrict__ qp,
                                        const float* __restrict__ kp,
                                        const float* __restrict__ b1,
                                        const float* __restrict__ W2,
                                        const float* __restrict__ b2,
                                        const unsigned char* __restrict__ mask,
                                        const float* __restrict__ v,
                                        float* __restrict__ attn_out,
                                        float* __restrict__ outA) {
  __shared__ float s_prob[NI];
  __shared__ float s_rmax[8];
  __shared__ float s_rsum[8];

  const int bo   = blockIdx.x;       // b*64 + o
  const int b    = bo >> 6;
  const int tid  = threadIdx.x;
  const int wave = tid >> 5;
  const int lane = tid & 31;

  // Per-lane slice of h: 8 contiguous values
  const int hbase = lane << 3;
  const float* qprow = qp + (size_t)bo * DIMD;
  float qv[8], wv[8];
#pragma unroll
  for (int j = 0; j < 8; ++j) {
    const int h = hbase + j;
    qv[j] = qprow[h] + b1[h];
    wv[j] = W2[h];
  }
  const float b2v = b2[0];
  const float* kpb = kp + (size_t)b * NI * DIMD;
  const unsigned char* mrow = mask + (size_t)b * NI;

  // Phase 1: each wave handles i = wave, wave+8, ...
  for (int i = wave; i < NI; i += 8) {
    const float* kprow = kpb + (size_t)i * DIMD + hbase;
    v4f ka = *(const v4f*)(kprow);
    v4f kb = *(const v4f*)(kprow + 4);
    float s = 0.0f;
#pragma unroll
    for (int j = 0; j < 4; ++j) s += wv[j] * fast_tanh(qv[j] + ka[j]);
#pragma unroll
    for (int j = 0; j < 4; ++j) s += wv[4 + j] * fast_tanh(qv[4 + j] + kb[j]);
#pragma unroll
    for (int off = 16; off > 0; off >>= 1) s += __shfl_xor(s, off, 32);
    if (lane == 0)
      s_prob[i] = mrow[i] ? -__builtin_inff() : (s + b2v);
  }
  __syncthreads();

  // Phase 2: softmax over 512 scores (2 per thread)
  const float e0 = s_prob[tid];
  const float e1 = s_prob[tid + 256];
  float mx = fmaxf(e0, e1);
#pragma unroll
  for (int off = 16; off > 0; off >>= 1) mx = fmaxf(mx, __shfl_xor(mx, off, 32));
  if (lane == 0) s_rmax[wave] = mx;
  __syncthreads();
  mx = s_rmax[0];
#pragma unroll
  for (int w = 1; w < 8; ++w) mx = fmaxf(mx, s_rmax[w]);

  float p0 = __expf(e0 - mx);
  float p1 = __expf(e1 - mx);
  float sm = p0 + p1;
#pragma unroll
  for (int off = 16; off > 0; off >>= 1) sm += __shfl_xor(sm, off, 32);
  if (lane == 0) s_rsum[wave] = sm;
  __syncthreads();
  sm = 0.0f;
#pragma unroll
  for (int w = 1; w < 8; ++w) sm += s_rsum[w];
  sm += s_rsum[0];
  const float inv = 1.0f / sm;
  p0 *= inv;
  p1 *= inv;
  s_prob[tid]       = p0;
  s_prob[tid + 256] = p1;
  attn_out[(size_t)bo * NI + tid]       = p0;
  attn_out[(size_t)bo * NI + tid + 256] = p1;
  __syncthreads();

  // Phase 3: outA[bo, d] = sum_i attn[i] * v[b, i, d]  (d = tid, coalesced)
  const float* vb = v + (size_t)b * NI * DIMD + tid;
  float acc = 0.0f;
#pragma unroll 8
  for (int i = 0; i < NI; ++i) acc += s_prob[i] * vb[(size_t)i * DIMD];
  outA[(size_t)bo * DIMD + tid] = acc;
}

extern "C" void kernel_launch(void* const* d_in, const int* in_sizes, int n_in,
                              void* d_out, int out_size, void* d_ws, size_t ws_size,
                              hipStream_t stream) {
  const float* q  = (const float*)d_in[0];
  const float* k  = (const float*)d_in[1];
  const float* v  = (const float*)d_in[2];
  const unsigned char* mask = (const unsigned char*)d_in[3];  // jax bool = 1 byte
  const float* W1 = (const float*)d_in[4];
  const float* b1 = (const float*)d_in[5];
  const float* W2 = (const float*)d_in[6];
  const float* b2 = (const float*)d_in[7];
  const float* Wf = (const float*)d_in[8];
  const float* bf = (const float*)d_in[9];

  float* out  = (float*)d_out;                 // (16,64,256) = 262144 floats
  float* attn = out + (size_t)NB * NO * DIMD;  // (16,64,512) = 524288 floats

  float* qp   = (float*)d_ws;                  // 1024*256
  float* kp   = qp + (size_t)NB * NO * DIMD;   // 8192*256
  float* outA = kp + (size_t)NB * NI * DIMD;   // 1024*256

  // qp[bo,h] = sum_d q[bo,d] * W1[h*512 + d]          (1024 rows -> 64 row-tiles)
  wmma_gemm_kernel<<<(64 * 16) / 8, 256, 0, stream>>>(q, W1, nullptr, qp, 512, 0, 64, 0);
  // kp[bi,h] = sum_d k[bi,d] * W1[h*512 + 256 + d]    (8192 rows -> 512 row-tiles)
  wmma_gemm_kernel<<<(512 * 16) / 8, 256, 0, stream>>>(k, W1, nullptr, kp, 512, 256, 512, 0);
  // fused score + softmax + attn@v: one block per (b,o)
  score_softmax_av_kernel<<<NB * NO, 256, 0, stream>>>(qp, kp, b1, W2, b2, mask, v, attn, outA);
  // out[bo,c] = leaky( sum_d outA[bo,d]*Wf[c*256+d] + bf[c] )
  wmma_gemm_kernel<<<(64 * 16) / 8, 256, 0, stream>>>(outA, Wf, bf, out, 256, 0, 64, 1);
}